// MyLSTM_43430709297747
// MI455X (gfx1250) — compile-verified
//
#include <hip/hip_runtime.h>

#define MC_ITER 10
#define NB 10   // batch
#define ND 3    // input dim
#define NH 4    // hidden dim

typedef float v2f __attribute__((ext_vector_type(2)));
typedef float v8f __attribute__((ext_vector_type(8)));

__device__ __forceinline__ float fsig(float x) {
  // 1/(1+e^-x) with native exp + rcp (short critical path)
  return __builtin_amdgcn_rcpf(1.0f + __expf(-x));
}
__device__ __forceinline__ float ftanh(float x) {
  float e = __expf(-2.0f * x);
  return (1.0f - e) * __builtin_amdgcn_rcpf(1.0f + e);
}

__global__ void mc_lstm_zero(float* __restrict__ p, int n) {
  int i = blockIdx.x * blockDim.x + threadIdx.x;
  if (i < n) p[i] = 0.0f;
}

// One wave (32 lanes) per MC-dropout iteration. Entire LSTM state lives in
// registers. Per timestep:
//   gates^T(16x16) = (W^T*zx)(16x4) @ x_t^T(4x16)  +  (U^T*zh)(16x4) @ h^T(4x16)
// done as two chained V_WMMA_F32_16X16X4_F32 (A matrices are loop-invariant).
// D layout: VGPR r holds channel r (lanes 0-15, N=batch) and channel r+8
// (lanes 16-31). With gate order i,f,o,g: acc[j] = {i_j | o_j},
// acc[j+4] = {f_j | g_j} -> the LSTM cell update needs only shfl_xor(16).
__global__ __launch_bounds__(32) void mc_lstm_kernel(
    const float* __restrict__ X,    // (T, B, D)
    const float* __restrict__ zx,   // (ITER, D)
    const float* __restrict__ zh,   // (ITER, H)
    const float* __restrict__ Wi, const float* __restrict__ Ui,
    const float* __restrict__ Wf, const float* __restrict__ Uf,
    const float* __restrict__ Wo, const float* __restrict__ Uo,
    const float* __restrict__ Wg, const float* __restrict__ Ug,
    float* __restrict__ out, int T) {
  const int lane = threadIdx.x;       // wave32
  const int it   = blockIdx.x;        // MC iteration
  const bool lo  = lane < 16;
  const int  m   = lane & 15;         // A-matrix row = gate channel
  const int  gate = m >> 2;           // 0:i 1:f 2:o 3:g
  const int  j    = m & 3;            // hidden unit within gate
  const int  kb   = lo ? 0 : 2;       // K pair held by this lane half

  const float* Utab[4] = {Ui, Uf, Uo, Ug};   // each (H,H) row-major
  const float* Wtab[4] = {Wi, Wf, Wo, Wg};   // each (D,H) row-major
  const float* Um = Utab[gate];
  const float* Wm = Wtab[gate];

  // Loop-invariant A matrices (16x4, f32): lane holds K=kb in .x, K=kb+1 in .y
  v2f Ua, Wa;
  Ua.x = Um[kb * NH + j]       * zh[it * NH + kb];
  Ua.y = Um[(kb + 1) * NH + j] * zh[it * NH + kb + 1];
  Wa.x = Wm[kb * NH + j]       * zx[it * ND + kb];                 // K=0 or 2
  Wa.y = lo ? (Wm[1 * NH + j]  * zx[it * ND + 1]) : 0.0f;          // K=1; K=3 pad

  // x_t^T B-matrix feed: lane<16 supplies x[t][b][0],x[t][b][1];
  // lane>=16 supplies x[t][b][2], 0. Padding batches (b>=NB) read offset 0
  // (values land in unused D columns).
  const int bt = lane & 15;
  int off  = (bt < NB) ? (bt * ND + (lo ? 0 : 2)) : 0;
  int offB = off + ((lo && bt < NB) ? 1 : 0);

  float hb0 = 0.0f, hb1 = 0.0f;   // h^T B-matrix registers
  float cst[4] = {0.f, 0.f, 0.f, 0.f};   // c state, lane = batch (lanes 0-15)

  const float inv = 1.0f / (float)MC_ITER;
  const size_t TBH = (size_t)T * NB * NH;
  float* out_o = out;
  float* out_h = out + TBH;
  float* out_c = out + 2 * TBH;

  for (int t = 0; t < T; ++t) {
    // x feed for this step (independent of the recurrence chain)
    float xv0 = X[off];
    float xr  = X[offB];
    float xv1 = lo ? xr : 0.0f;
    off  += NB * ND;
    offB += NB * ND;

    v2f xb;  xb.x  = xv0; xb.y  = xv1;
    v2f hbv; hbv.x = hb0; hbv.y = hb1;

    v8f acc = {};
    acc = __builtin_amdgcn_wmma_f32_16x16x4_f32(
        false, Wa, false, xb, (short)0, acc, false, false);
    acc = __builtin_amdgcn_wmma_f32_16x16x4_f32(
        false, Ua, false, hbv, (short)0, acc, false, false);

    float h2v[4];
#pragma unroll
    for (int jj = 0; jj < 4; ++jj) {
      float va = acc[jj];       // lanes0-15: i_j[b]   lanes16-31: o_j[b]
      float vb = acc[jj + 4];   // lanes0-15: f_j[b]   lanes16-31: g_j[b]
      float sa = fsig(va);
      float sb = fsig(vb);
      float tb = ftanh(vb);
      float gg = __shfl_xor(tb, 16, 32);   // g_j[b] -> lanes 0-15
      float oo = __shfl_xor(sa, 16, 32);   // o_j[b] -> lanes 0-15
      float cn = sb * cst[jj] + sa * gg;   // f*c + i*g   (lanes 0-15 valid)
      float hn = oo * ftanh(cn);           // h2          (lanes 0-15 valid)
      cst[jj] = cn;
      h2v[jj] = hn;
      if (lane < NB) {                     // fire-and-forget mean accumulate
        size_t idx = ((size_t)t * NB + lane) * NH + jj;
        unsafeAtomicAdd(out_o + idx, oo * inv);
        unsafeAtomicAdd(out_h + idx, hn * inv);
        unsafeAtomicAdd(out_c + idx, cn * inv);
      }
    }
    // Rebuild h^T B-matrix for next step: lanes>=16 need h2[b][2], h2[b][3]
    float hi0 = __shfl_xor(h2v[2], 16, 32);
    float hi1 = __shfl_xor(h2v[3], 16, 32);
    hb0 = lo ? h2v[0] : hi0;
    hb1 = lo ? h2v[1] : hi1;
  }
}

extern "C" void kernel_launch(void* const* d_in, const int* in_sizes, int n_in,
                              void* d_out, int out_size, void* d_ws, size_t ws_size,
                              hipStream_t stream) {
  const float* X  = (const float*)d_in[0];
  const float* zx = (const float*)d_in[1];
  const float* zh = (const float*)d_in[2];
  const float* Wi = (const float*)d_in[3];
  const float* Ui = (const float*)d_in[4];
  const float* Wf = (const float*)d_in[5];
  const float* Uf = (const float*)d_in[6];
  const float* Wo = (const float*)d_in[7];
  const float* Uo = (const float*)d_in[8];
  const float* Wg = (const float*)d_in[9];
  const float* Ug = (const float*)d_in[10];
  float* out = (float*)d_out;

  const int T = in_sizes[0] / (NB * ND);   // 16384

  mc_lstm_zero<<<(out_size + 255) / 256, 256, 0, stream>>>(out, out_size);
  mc_lstm_kernel<<<MC_ITER, 32, 0, stream>>>(X, zx, zh, Wi, Ui, Wf, Uf,
                                             Wo, Uo, Wg, Ug, out, T);
}